// myRecogniton_57707180589241
// MI455X (gfx1250) — compile-verified
//
#include <hip/hip_runtime.h>
#include <hip/hip_bf16.h>
#include <math.h>

// ---------------------------------------------------------------------------
// MI455X (gfx1250, wave32) implementation.
// FP32 WMMA (v_wmma_f32_16x16x4_f32) everywhere: workload is ~10 GFLOP with a
// ~20MB working set (fully L2-resident at 192MB), dominated by the serial
// 64-step LSTM recurrence and ~100M tanh ops in the fused attention — matrix
// precision/throughput is not the bottleneck, so exact f32 semantics win.
// ---------------------------------------------------------------------------

typedef __attribute__((ext_vector_type(2))) float v2f;
typedef __attribute__((ext_vector_type(8))) float v8f;

#define TT 64      // time steps (W)
#define BB 16      // batch
#define CC 256     // conv channels / LSTM input
#define HD 256     // hidden dim
#define G4 1024    // 4*HD
#define EE 256     // embedding
#define HWN 384    // 6*64 spatial positions
#define KC9 2304   // C*9 for 3x3 conv

// ---------------------------------------------------------------------------
// Kernel 1: max over H (MaxPool2d((6,1))) + transpose -> x[t][b][c], which is
// exactly the row-major A matrix [M=T*B=1024, K=C=256] for the input GEMMs.
// ---------------------------------------------------------------------------
__global__ void k_maxpool(const float* __restrict__ conv_f, float* __restrict__ x) {
    int i = blockIdx.x * blockDim.x + threadIdx.x;   // 0 .. T*B*C-1 = 262143
    int c = i & 255;
    int b = (i >> 8) & 15;
    int t = i >> 12;
    const float* p = conv_f + ((b * CC + c) * 6) * 64 + t;
    float m = p[0];
#pragma unroll
    for (int h = 1; h < 6; ++h) m = fmaxf(m, p[h * 64]);
    x[i] = m;                                        // x[(t*16+b)*256 + c]
}

// ---------------------------------------------------------------------------
// Kernel 2: generic FP32 WMMA GEMM:  Out[M,N] = A[M,K] * W[N,K]^T + bias[N]
// One wave per 16(M) x 32(N) tile pair: the A fragment is loaded once and
// reused by two WMMAs (halves A traffic, 3 loads : 2 WMMAs per K-step).
// Fragment layouts per CDNA5 ISA 7.12.2:
//   A frag a[v]  = A[m0 + lane%16][k0 + 2*(lane/16) + v]
//   B frag b[v]  = B[k0 + 2*(lane/16) + v][n + lane%16] = W[n][k]
//   D frag d[v] -> (m = m0 + v + 8*(lane/16), n = n0 + lane%16)
// M mult of 16, N mult of 32, K mult of 4. Guard is wave-uniform.
// ---------------------------------------------------------------------------
__global__ void k_gemm_wmma(const float* __restrict__ A, const float* __restrict__ W,
                            const float* __restrict__ bias, float* __restrict__ Out,
                            int M, int N, int K) {
    int wave = blockIdx.x * (blockDim.x >> 5) + (threadIdx.x >> 5);
    int lane = threadIdx.x & 31;
    int npair = N >> 5;                              // pairs of 16-wide N tiles
    int total = (M >> 4) * npair;
    if (wave >= total) return;                       // uniform per wave
    int m0 = (wave / npair) << 4;
    int n0 = (wave % npair) << 5;
    int lm = lane & 15;
    int lh = lane >> 4;                              // lane half: K sub-offset
    const float* arow  = A + (size_t)(m0 + lm) * K + 2 * lh;
    const float* wrow0 = W + (size_t)(n0 + lm) * K + 2 * lh;
    const float* wrow1 = W + (size_t)(n0 + 16 + lm) * K + 2 * lh;
    v8f acc0 = {}, acc1 = {};
    for (int k0 = 0; k0 < K; k0 += 4) {
        v2f a, b0, b1;
        a[0]  = arow[k0];  a[1]  = arow[k0 + 1];
        b0[0] = wrow0[k0]; b0[1] = wrow0[k0 + 1];
        b1[0] = wrow1[k0]; b1[1] = wrow1[k0 + 1];
        acc0 = __builtin_amdgcn_wmma_f32_16x16x4_f32(false, a, false, b0,
                                                     (short)0, acc0, false, false);
        acc1 = __builtin_amdgcn_wmma_f32_16x16x4_f32(false, a, false, b1,
                                                     (short)0, acc1, false, false);
    }
    float bv0 = bias ? bias[n0 + lm] : 0.0f;
    float bv1 = bias ? bias[n0 + 16 + lm] : 0.0f;
#pragma unroll
    for (int v = 0; v < 8; ++v) {
        int m = m0 + v + 8 * lh;
        Out[(size_t)m * N + n0 + lm]      = acc0[v] + bv0;
        Out[(size_t)m * N + n0 + 16 + lm] = acc1[v] + bv1;
    }
}

// ---------------------------------------------------------------------------
// Kernel 3: bidirectional LSTM recurrence. grid = {2 directions}, 512 thr =
// 16 waves. Wave w owns hidden columns [16w, 16w+16) and computes all four
// gate tiles (i/f/g/o differ only by a 256-column offset), so the nonlinear
// cell update happens entirely in registers; cell state c never leaves VGPRs.
// K-loop is outermost so the h fragment is read from LDS once per K-step and
// reused by 4 independent WMMAs (no RAW hazard between them). Only h (16KB)
// lives in LDS, with 2 barriers per time step.
// ---------------------------------------------------------------------------
__global__ void k_lstm(const float* __restrict__ xp_f, const float* __restrict__ xp_b,
                       const float* __restrict__ Whh_f, const float* __restrict__ Whh_b,
                       float* __restrict__ enc) {
    __shared__ float h_l[BB * HD];                   // 16 KB
    int dir = blockIdx.x;
    const float* xp  = dir ? xp_b  : xp_f;
    const float* Whh = dir ? Whh_b : Whh_f;
    int tid = threadIdx.x;
    int lane = tid & 31, wv = tid >> 5;
    int lm = lane & 15, lh = lane >> 4;
    int j0 = wv << 4;                                // this wave's column tile

    float cst[8];
#pragma unroll
    for (int v = 0; v < 8; ++v) cst[v] = 0.0f;
    for (int i = tid; i < BB * HD; i += 512) h_l[i] = 0.0f;
    __syncthreads();

    const float* hrow = h_l + lm * HD + 2 * lh;      // A-frag base (m=lm)
    const float* wr[4];
#pragma unroll
    for (int gi = 0; gi < 4; ++gi)
        wr[gi] = Whh + (size_t)(gi * HD + j0 + lm) * HD + 2 * lh;

    for (int tt = 0; tt < TT; ++tt) {
        int t = dir ? (TT - 1 - tt) : tt;            // reverse scan keeps time order
        const float* xpt = xp + (size_t)t * (BB * G4);

        v8f acc[4];
        v8f zero = {};
#pragma unroll
        for (int gi = 0; gi < 4; ++gi) acc[gi] = zero;

        for (int k0 = 0; k0 < HD; k0 += 4) {
            v2f a;
            a[0] = hrow[k0]; a[1] = hrow[k0 + 1];    // one LDS read, 4 WMMAs
#pragma unroll
            for (int gi = 0; gi < 4; ++gi) {
                v2f b;
                b[0] = wr[gi][k0]; b[1] = wr[gi][k0 + 1];
                acc[gi] = __builtin_amdgcn_wmma_f32_16x16x4_f32(false, a, false, b,
                                                                (short)0, acc[gi],
                                                                false, false);
            }
        }
        __syncthreads();                             // all h_l reads complete
#pragma unroll
        for (int v = 0; v < 8; ++v) {
            int m = v + 8 * lh;                      // batch index
            int j = j0 + lm;                         // hidden column
            float gi_ = acc[0][v] + xpt[m * G4 + j];
            float gf_ = acc[1][v] + xpt[m * G4 + HD + j];
            float gg_ = acc[2][v] + xpt[m * G4 + 2 * HD + j];
            float go_ = acc[3][v] + xpt[m * G4 + 3 * HD + j];
            float si = 1.0f / (1.0f + expf(-gi_));
            float sf = 1.0f / (1.0f + expf(-gf_));
            float so = 1.0f / (1.0f + expf(-go_));
            float c  = sf * cst[v] + si * tanhf(gg_);
            cst[v] = c;
            float h = so * tanhf(c);
            h_l[m * HD + j] = h;
            enc[(size_t)(t * BB + m) * (2 * HD) + dir * HD + j] = h;
        }
        __syncthreads();                             // h_l ready for next step
    }
}

// ---------------------------------------------------------------------------
// Kernel 4: attention conv 3x3 SAME as implicit GEMM with WMMA.
// M = B*H*W = 6144 pixels, N = E = 256, K = C*9 = 2304. A fragments gathered
// with zero padding (divergent loads reconverge before the WMMA issues).
// Output stored directly as xe[b][n=h*64+w][e]  ([B, HW, E]).
// ---------------------------------------------------------------------------
__global__ void k_attconv(const float* __restrict__ conv_f, const float* __restrict__ Wc,
                          const float* __restrict__ bc, float* __restrict__ xe) {
    int wave = blockIdx.x * (blockDim.x >> 5) + (threadIdx.x >> 5);
    int lane = threadIdx.x & 31;
    const int ntiles = EE >> 4;                      // 16
    if (wave >= 384 * ntiles) return;                // 6144 tiles, uniform
    int m0 = (wave / ntiles) << 4;
    int n0 = (wave % ntiles) << 4;
    int lm = lane & 15, lh = lane >> 4;

    int p   = m0 + lm;                               // pixel row of A
    int b   = p / HWN;
    int rem = p - b * HWN;
    int y   = rem >> 6;
    int xpix = rem & 63;

    const float* wrow = Wc + (size_t)(n0 + lm) * KC9 + 2 * lh;
    v8f acc = {};
    for (int k0 = 0; k0 < KC9; k0 += 4) {
        v2f av, bv;
#pragma unroll
        for (int v = 0; v < 2; ++v) {
            int r   = k0 + 2 * lh + v;
            int c   = r / 9;
            int tap = r - c * 9;
            int dy  = tap / 3 - 1;
            int dx  = (tap - (tap / 3) * 3) - 1;
            int yy = y + dy, xx = xpix + dx;
            float val = 0.0f;
            if ((unsigned)yy < 6u && (unsigned)xx < 64u)
                val = conv_f[((size_t)(b * CC + c) * 6 + yy) * 64 + xx];
            av[v] = val;
            bv[v] = wrow[k0 + v];
        }
        acc = __builtin_amdgcn_wmma_f32_16x16x4_f32(false, av, false, bv,
                                                    (short)0, acc, false, false);
    }
    float bias = bc[n0 + lm];
#pragma unroll
    for (int v = 0; v < 8; ++v) {
        int pp = m0 + v + 8 * lh;                    // pixel = (b, n)
        xe[(size_t)pp * EE + n0 + lm] = acc[v] + bias;
    }
}

// ---------------------------------------------------------------------------
// Kernel 5: fused attention per (t, b): scores e[n] = w . tanh(xe[b,n,:] + g)
// + vb, softmax over n (384), context att[c] = sum_n alpha[n]*conv_f[b,c,n],
// and the final concat/transpose write. Avoids the 400MB [T,B,HW,E] tensor.
// Rows are 16B-aligned (ws offsets are multiples of 2^18 floats), so float4
// loads (global_load_b128) quarter VMEM issue in the hottest loop.
// ---------------------------------------------------------------------------
__global__ void k_attention(const float* __restrict__ xe, const float* __restrict__ g,
                            const float* __restrict__ hidden_de,
                            const float* __restrict__ conv_f,
                            const float* __restrict__ attv_w,
                            const float* __restrict__ attv_b,
                            float* __restrict__ out) {
    __shared__ float g_l[EE];
    __shared__ float w_l[EE];
    __shared__ float e_l[HWN];
    __shared__ float red[256];
    int tb = blockIdx.x;                             // t*16 + b
    int t = tb >> 4;
    int b = tb & 15;
    int tid = threadIdx.x;                           // 256 threads

    g_l[tid] = g[(size_t)tb * EE + tid];
    w_l[tid] = attv_w[tid];
    float vb = attv_b[0];
    __syncthreads();

    for (int n = tid; n < HWN; n += 256) {
        const float4* xr4 =
            reinterpret_cast<const float4*>(xe + (size_t)(b * HWN + n) * EE);
        float s = vb;
        for (int k4 = 0; k4 < EE / 4; ++k4) {
            float4 xv = xr4[k4];
            int k = k4 * 4;
            s += w_l[k]     * tanhf(xv.x + g_l[k]);
            s += w_l[k + 1] * tanhf(xv.y + g_l[k + 1]);
            s += w_l[k + 2] * tanhf(xv.z + g_l[k + 2]);
            s += w_l[k + 3] * tanhf(xv.w + g_l[k + 3]);
        }
        e_l[n] = s;
    }
    __syncthreads();

    // softmax max
    float lmx = -3.4e38f;
    for (int n = tid; n < HWN; n += 256) lmx = fmaxf(lmx, e_l[n]);
    red[tid] = lmx;
    __syncthreads();
    for (int s = 128; s > 0; s >>= 1) {
        if (tid < s) red[tid] = fmaxf(red[tid], red[tid + s]);
        __syncthreads();
    }
    float mx = red[0];
    __syncthreads();

    // exp + sum
    float ls = 0.0f;
    for (int n = tid; n < HWN; n += 256) {
        float ex = expf(e_l[n] - mx);
        e_l[n] = ex;
        ls += ex;
    }
    red[tid] = ls;
    __syncthreads();
    for (int s = 128; s > 0; s >>= 1) {
        if (tid < s) red[tid] += red[tid + s];
        __syncthreads();
    }
    float inv = 1.0f / red[0];
    __syncthreads();
    for (int n = tid; n < HWN; n += 256) e_l[n] *= inv;
    __syncthreads();

    // context: thread 'tid' = channel c; conv_f rows are 1536B -> 16B aligned
    const float4* or4 =
        reinterpret_cast<const float4*>(conv_f + (size_t)(b * CC + tid) * HWN);
    float s = 0.0f;
    for (int n4 = 0; n4 < HWN / 4; ++n4) {
        float4 ov = or4[n4];
        int n = n4 * 4;
        s += e_l[n] * ov.x + e_l[n + 1] * ov.y + e_l[n + 2] * ov.z + e_l[n + 3] * ov.w;
    }

    float* orow_out = out + (size_t)(b * TT + t) * (HD + CC);   // out[b][t][:]
    orow_out[tid]      = hidden_de[(size_t)tb * HD + tid];
    orow_out[HD + tid] = s;
}

// ---------------------------------------------------------------------------
// Launch
// ---------------------------------------------------------------------------
extern "C" void kernel_launch(void* const* d_in, const int* in_sizes, int n_in,
                              void* d_out, int out_size, void* d_ws, size_t ws_size,
                              hipStream_t stream) {
    const float* conv_f = (const float*)d_in[0];
    const float* Wih_f  = (const float*)d_in[1];
    const float* Whh_f  = (const float*)d_in[2];
    const float* b_f    = (const float*)d_in[3];
    const float* Wih_b  = (const float*)d_in[4];
    const float* Whh_b  = (const float*)d_in[5];
    const float* b_b    = (const float*)d_in[6];
    const float* dec_W  = (const float*)d_in[7];
    const float* dec_b  = (const float*)d_in[8];
    const float* atth_W = (const float*)d_in[9];
    const float* atth_b = (const float*)d_in[10];
    const float* attc_W = (const float*)d_in[11];
    const float* attc_b = (const float*)d_in[12];
    const float* attv_w = (const float*)d_in[13];
    const float* attv_b = (const float*)d_in[14];
    float* out = (float*)d_out;

    float* ws    = (float*)d_ws;
    float* xbuf  = ws;                   // [1024, 256]          262144
    float* xp_f  = xbuf + 262144;        // [64, 16, 1024]       1048576
    float* xp_b  = xp_f + 1048576;       // [64, 16, 1024]       1048576
    float* enc   = xp_b + 1048576;       // [64, 16, 512]        524288
    float* hde   = enc  + 524288;        // [64, 16, 256]        262144
    float* gbuf  = hde  + 262144;        // [64, 16, 256]        262144
    float* xebuf = gbuf + 262144;        // [16, 384, 256]       1572864

    // 1. maxpool + transpose
    k_maxpool<<<1024, 256, 0, stream>>>(conv_f, xbuf);
    // 2. xp = x @ Wih^T + b  (M=1024,N=1024,K=256): 64*32 = 2048 waves
    k_gemm_wmma<<<256, 256, 0, stream>>>(xbuf, Wih_f, b_f, xp_f, 1024, 1024, 256);
    k_gemm_wmma<<<256, 256, 0, stream>>>(xbuf, Wih_b, b_b, xp_b, 1024, 1024, 256);
    // 3. recurrence (both directions in parallel)
    k_lstm<<<2, 512, 0, stream>>>(xp_f, xp_b, Whh_f, Whh_b, enc);
    // 4. hidden_de = enc @ dec_W^T + dec_b  (M=1024,N=256,K=512): 512 waves
    k_gemm_wmma<<<64, 256, 0, stream>>>(enc, dec_W, dec_b, hde, 1024, 256, 512);
    // 5. g = hidden_de @ atth_W^T + atth_b  (M=1024,N=256,K=256): 512 waves
    k_gemm_wmma<<<64, 256, 0, stream>>>(hde, atth_W, atth_b, gbuf, 1024, 256, 256);
    // 6. attention conv (implicit GEMM): 6144 tiles / 8 waves
    k_attconv<<<768, 256, 0, stream>>>(conv_f, attc_W, attc_b, xebuf);
    // 7. fused attention + output concat/transpose
    k_attention<<<1024, 256, 0, stream>>>(xebuf, gbuf, hde, conv_f, attv_w, attv_b, out);
}